// BoundaryLoss_14285061226976
// MI455X (gfx1250) — compile-verified
//
#include <hip/hip_runtime.h>

// BoundaryLoss for MI455X / gfx1250 (wave32).
//  Pass 1: O(H) two-scan vertical nearest-feature distance per column (u16).
//  Pass 2: O(W^2) min-plus parabola envelope per row from LDS (rows staged
//          via GLOBAL_LOAD_ASYNC_TO_LDS when available), fused with channel
//          softmax and the weighted-mean reduction.
//  Reductions use V_WMMA_F32_16X16X4_F32 with an all-ones B matrix (exact
//  f32 FMA accumulate on the matrix pipe); no atomics -> bitwise deterministic.

#define Bd 4
#define Cd 4
#define Hd 256
#define Wd 256
#define NTOT (Bd * Cd * Hd * Wd)
#define DCAP 1023            // sentinel distance: DCAP^2 = 1046529 > THRESH
#define THRESH 1.0e6f        // real max h = 2*255^2 = 130050 << THRESH

#if defined(__has_builtin)
#if __has_builtin(__builtin_amdgcn_global_load_async_to_lds_b32)
#define HAVE_ASYNC_LDS 1
#endif
#endif
#ifndef HAVE_ASYNC_LDS
#define HAVE_ASYNC_LDS 0
#endif

typedef __attribute__((ext_vector_type(2))) float v2f;
typedef __attribute__((ext_vector_type(8))) float v8f;

#if HAVE_ASYNC_LDS
typedef __attribute__((address_space(1))) int as1_int;   // global (__device__)
typedef __attribute__((address_space(3))) int as3_int;   // LDS (__shared__)
#endif

__device__ __forceinline__ int imin(int a, int b) { return a < b ? a : b; }

#if HAVE_ASYNC_LDS
__device__ __forceinline__ void wait_asynccnt0() {
#if __has_builtin(__builtin_amdgcn_s_wait_asynccnt)
  __builtin_amdgcn_s_wait_asynccnt(0);
#else
  asm volatile("s_wait_asynccnt 0x0" ::: "memory");
#endif
}
#endif

// Wave32 sum via WMMA ones-trick.
// A (16x4 f32): lane L holds (A[M, k0], A[M, k0+1]) with M=L%16, k0 = (L<16)?0:2.
// With a = {val, 0} and B = all-ones: D[m,n] = val_m + val_{m+16}.
// Per-lane Sum(d[0..7]) covers m=0..7 (lanes 0-15) or m=8..15 (lanes 16-31),
// so lane0 + lane16 = full 32-lane sum. Exact f32 FMA -> same math as tree add.
__device__ __forceinline__ float wave_sum_wmma(float v) {
  v2f a; a[0] = v;    a[1] = 0.0f;
  v2f b; b[0] = 1.0f; b[1] = 1.0f;
  v8f c = {};
  v8f d = __builtin_amdgcn_wmma_f32_16x16x4_f32(false, a, false, b,
                                                (short)0, c, false, false);
  float s = d[0] + d[1] + d[2] + d[3] + d[4] + d[5] + d[6] + d[7];
  return __shfl(s, 0, 32) + __shfl(s, 16, 32);
}

// ---------------- Pass 1: vertical nearest-feature distance per column ------
// One block per (b,c) channel, one thread per column x. Two scans (down, up).
// Also computes has_fg per channel.
__global__ void col_pass(const float* __restrict__ target,
                         unsigned short* __restrict__ gfg,
                         unsigned short* __restrict__ gbg,
                         int* __restrict__ hasfg) {
  const int bc = blockIdx.x;          // 0 .. B*C-1
  const int x  = threadIdx.x;         // 0 .. W-1
  const size_t base = (size_t)bc * Hd * Wd + (size_t)x;

  int dfg = DCAP, dbg = DCAP, any = 0;
  for (int y = 0; y < Hd; ++y) {      // downward scan (coalesced across x)
    const size_t i = base + (size_t)y * Wd;
    const bool fg = target[i] > 0.5f;
    dfg = fg ? 0 : imin(dfg + 1, DCAP);
    dbg = fg ? imin(dbg + 1, DCAP) : 0;
    any |= (int)fg;
    gfg[i] = (unsigned short)dfg;
    gbg[i] = (unsigned short)dbg;
  }
  dfg = DCAP; dbg = DCAP;
  for (int y = Hd - 1; y >= 0; --y) { // upward scan, combine with min
    const size_t i = base + (size_t)y * Wd;
    const int vf = (int)gfg[i];
    const int vb = (int)gbg[i];
    const bool fg = (vf == 0);
    dfg = fg ? 0 : imin(dfg + 1, DCAP);
    dbg = fg ? imin(dbg + 1, DCAP) : 0;
    gfg[i] = (unsigned short)imin(vf, dfg);
    gbg[i] = (unsigned short)imin(vb, dbg);
  }

  __shared__ int sany[Wd];
  sany[x] = any;
  __syncthreads();
  for (int off = Wd / 2; off > 0; off >>= 1) {
    if (x < off) sany[x] |= sany[x + off];
    __syncthreads();
  }
  if (x == 0) hasfg[bc] = sany[0];
}

// ---------------- Pass 2: row min-plus + softmax + weighted partial sum -----
// One block per (b,c,y) row, one thread per x. g row staged in LDS (async DMA
// when available; broadcast reads in the envelope loop), fused softmax over
// C=4, WMMA wave reduction, one partial per block.
__global__ void row_pass(const float* __restrict__ pred,
                         const unsigned short* __restrict__ gfg,
                         const unsigned short* __restrict__ gbg,
                         const int* __restrict__ hasfg,
                         float* __restrict__ partials) {
  const int row = blockIdx.x;         // 0 .. B*C*H-1
  const int y   = row % Hd;
  const int bc  = row / Hd;
  const int c   = bc % Cd;
  const int b   = bc / Cd;
  const int x   = threadIdx.x;

  const size_t hw    = (size_t)Hd * Wd;
  const size_t pbase = (((size_t)b * Cd) * Hd + (size_t)y) * Wd + (size_t)x;
  __builtin_prefetch(&pred[pbase], 0, 1);           // -> global_prefetch_b8
  __builtin_prefetch(&pred[pbase + hw], 0, 1);
  __builtin_prefetch(&pred[pbase + 2 * hw], 0, 1);
  __builtin_prefetch(&pred[pbase + 3 * hw], 0, 1);

  __shared__ float sfg[Wd];
  __shared__ float sbg[Wd];
  const size_t gbase = (size_t)bc * hw + (size_t)y * Wd;

#if HAVE_ASYNC_LDS
  // Stage both u16 rows (2 x 512 B) into LDS with the async DMA engine:
  // 256 lanes x b32 = 1024 B. Threads 0-127 carry the fg row, 128-255 the bg
  // row. Row bases are 512-B aligned and per-lane offsets are 4*t, so every
  // b32 transfer is 4-B aligned. Per-wave ASYNCcnt wait + workgroup barrier
  // makes all waves' DMA'd data visible.
  __shared__ unsigned short stage[2 * Wd];
  {
    const int t = x;
    const unsigned short* gsrc =
        (t < 128) ? (gfg + gbase + 2 * (size_t)t)
                  : (gbg + gbase + 2 * (size_t)(t - 128));
    unsigned short* ldst =
        (t < 128) ? (stage + 2 * t) : (stage + Wd + 2 * (t - 128));
    __builtin_amdgcn_global_load_async_to_lds_b32(
        (as1_int*)(void*)const_cast<unsigned short*>(gsrc),
        (as3_int*)(void*)ldst, 0, 0);
    wait_asynccnt0();
  }
  __syncthreads();
  {
    const float df = (float)stage[x];
    const float db = (float)stage[Wd + x];
    sfg[x] = df * df;                 // squared vertical distance (or sentinel)
    sbg[x] = db * db;
  }
#else
  {
    const float df = (float)gfg[gbase + x];
    const float db = (float)gbg[gbase + x];
    sfg[x] = df * df;
    sbg[x] = db * db;
  }
#endif
  __syncthreads();

  float hfg = 3.0e37f, hbg = 3.0e37f;
  float diff = (float)x;              // (x - xp), decremented each iteration
#pragma unroll 8
  for (int xp = 0; xp < Wd; ++xp) {
    const float gf = sfg[xp];         // LDS broadcast
    const float gb = sbg[xp];
    hfg = fminf(hfg, __builtin_fmaf(diff, diff, gf));
    hbg = fminf(hbg, __builtin_fmaf(diff, diff, gb));
    diff -= 1.0f;
  }
  // _edt clamp: no feature anywhere in channel -> h >= DCAP^2 > THRESH -> 0
  const float efg  = (hfg > THRESH) ? 0.0f : __builtin_sqrtf(hfg);
  const float ebg  = (hbg > THRESH) ? 0.0f : __builtin_sqrtf(hbg);
  const float dist = (efg + ebg) * (hasfg[bc] ? 1.0f : 0.0f);

  // softmax over C = 4
  const float v0 = pred[pbase];
  const float v1 = pred[pbase + hw];
  const float v2 = pred[pbase + 2 * hw];
  const float v3 = pred[pbase + 3 * hw];
  const float m  = fmaxf(fmaxf(v0, v1), fmaxf(v2, v3));
  const float e0 = __expf(v0 - m), e1 = __expf(v1 - m);
  const float e2 = __expf(v2 - m), e3 = __expf(v3 - m);
  const float inv = 1.0f / (e0 + e1 + e2 + e3);
  const float ec  = (c == 0) ? e0 : (c == 1) ? e1 : (c == 2) ? e2 : e3;
  const float val = ec * inv * dist;

  const float wsum = wave_sum_wmma(val);   // uniform flow here: EXEC all ones
  __shared__ float wacc[8];
  const int wave = x >> 5, lane = x & 31;
  if (lane == 0) wacc[wave] = wsum;
  __syncthreads();
  if (x == 0) {
    float s = 0.0f;
    for (int i = 0; i < 8; ++i) s += wacc[i];   // fixed order -> deterministic
    partials[row] = s;
  }
}

// ---------------- Final deterministic reduction -----------------------------
__global__ void final_reduce(const float* __restrict__ partials,
                             float* __restrict__ out, int n, float scale) {
  const int t = threadIdx.x;
  float s = 0.0f;
  for (int i = t; i < n; i += 256) s += partials[i];  // fixed strided order
  const float wsum = wave_sum_wmma(s);
  __shared__ float wacc[8];
  const int wave = t >> 5, lane = t & 31;
  if (lane == 0) wacc[wave] = wsum;
  __syncthreads();
  if (t == 0) {
    float tot = 0.0f;
    for (int i = 0; i < 8; ++i) tot += wacc[i];
    out[0] = tot * scale;
  }
}

extern "C" void kernel_launch(void* const* d_in, const int* in_sizes, int n_in,
                              void* d_out, int out_size, void* d_ws, size_t ws_size,
                              hipStream_t stream) {
  (void)in_sizes; (void)n_in; (void)out_size; (void)ws_size;
  const float* pred   = (const float*)d_in[0];
  const float* target = (const float*)d_in[1];

  // Workspace layout (~4.02 MB): u16 gfg[N], u16 gbg[N], int hasfg[16],
  // float partials[B*C*H].
  unsigned short* gfg = (unsigned short*)d_ws;
  unsigned short* gbg = gfg + NTOT;
  int* hasfg          = (int*)(gbg + NTOT);          // 4 MB offset, aligned
  float* partials     = (float*)(hasfg + Bd * Cd);

  col_pass<<<dim3(Bd * Cd), dim3(Wd), 0, stream>>>(target, gfg, gbg, hasfg);
  row_pass<<<dim3(Bd * Cd * Hd), dim3(Wd), 0, stream>>>(pred, gfg, gbg, hasfg,
                                                        partials);
  final_reduce<<<dim3(1), dim3(256), 0, stream>>>(partials, (float*)d_out,
                                                  Bd * Cd * Hd,
                                                  1.0f / (float)NTOT);
}